// GNNLayer_49795850829975
// MI455X (gfx1250) — compile-verified
//
#include <hip/hip_runtime.h>

typedef float v2f __attribute__((ext_vector_type(2)));
typedef float v8f __attribute__((ext_vector_type(8)));

#define LEAKY 0.01f
#define IN_DIM 64
#define OUT_DIM 64
#define EF_DIM 32
#define KE (2 * IN_DIM + EF_DIM)   // 160
#define KEP (KE + 1)               // 161 (odd stride -> conflict-free LDS)
#define KN (OUT_DIM + IN_DIM)      // 128
#define KNP (KN + 1)               // 129

__device__ __forceinline__ float leaky(float x) {
    return x > 0.0f ? x : x * LEAKY;
}

__global__ void zero_f32(float* __restrict__ p, int n) {
    int i = blockIdx.x * blockDim.x + threadIdx.x;
    int stride = gridDim.x * blockDim.x;
    for (; i < n; i += stride) p[i] = 0.0f;
}

// msg = LeakyReLU([nf[src] | nf[dst] | ef] @ We^T); atomically reduce at dst.
// Block = 128 threads = 4 waves; each wave owns 16 output columns.
__global__ void __launch_bounds__(128) edge_msg_wmma(
    const float* __restrict__ nf, const float* __restrict__ ef,
    const int* __restrict__ src, const int* __restrict__ dst,
    const float* __restrict__ We, float* __restrict__ red,
    int E, int numTiles)
{
    __shared__ float sWe[OUT_DIM * KEP];  // ~41 KB
    __shared__ float sA[16 * KEP];        // ~10 KB
    __shared__ int sSrc[16];
    __shared__ int sDst[16];

    const int tid  = threadIdx.x;
    const int lane = tid & 31;
    const int wave = tid >> 5;
    const int E2   = 2 * E;

    // stage We once per block (row-major [64][160], padded stride)
    for (int i = tid; i < OUT_DIM * KE; i += 128) {
        int r = i / KE, c = i - r * KE;
        sWe[r * KEP + c] = We[i];
    }

    const int row   = lane & 15;          // A: M index
    const int kb    = (lane >> 4) * 2;    // A/B: K sub-offset {0,2}
    const int col   = (lane & 15) + wave * 16;  // B/D: N index
    const int mbase = (lane >> 4) * 8;    // D: M base {0,8}

    for (int tile = blockIdx.x; tile < numTiles; tile += gridDim.x) {
        const int e0 = tile * 16;
        __syncthreads();  // protect LDS from previous iteration's readers
        if (tid < 16) {
            int e = e0 + tid;
            int s = 0, d = 0;
            if (e < E)       { s = src[e];     d = dst[e];     }  // forward
            else if (e < E2) { s = dst[e - E]; d = src[e - E]; }  // reverse
            sSrc[tid] = s;
            sDst[tid] = d;
        }
        __syncthreads();
        // gather A tile: 16 edges x 160 features
        for (int i = tid; i < 16 * KE; i += 128) {
            int r = i / KE, c = i - r * KE;
            int e = e0 + r;
            float v = 0.0f;
            if (e < E2) {
                if (c < IN_DIM)          v = nf[sSrc[r] * IN_DIM + c];
                else if (c < 2 * IN_DIM) v = nf[sDst[r] * IN_DIM + (c - IN_DIM)];
                else {
                    int eb = (e < E) ? e : (e - E);
                    v = ef[eb * EF_DIM + (c - 2 * IN_DIM)];
                }
            }
            sA[r * KEP + c] = v;
        }
        __syncthreads();

        v8f acc = {0.f, 0.f, 0.f, 0.f, 0.f, 0.f, 0.f, 0.f};
        #pragma unroll
        for (int k = 0; k < KE; k += 4) {
            v2f a, b;
            a.x = sA[row * KEP + k + kb];
            a.y = sA[row * KEP + k + kb + 1];
            b.x = sWe[col * KEP + k + kb];
            b.y = sWe[col * KEP + k + kb + 1];
            acc = __builtin_amdgcn_wmma_f32_16x16x4_f32(
                false, a, false, b, (short)0, acc, false, false);
        }

        // LeakyReLU + scatter-add (f32 hw atomic, resolves in L2)
        #pragma unroll
        for (int i = 0; i < 8; ++i) {
            int m = mbase + i;
            float v = leaky(acc[i]);
            unsafeAtomicAdd(&red[sDst[m] * OUT_DIM + col], v);
        }
    }
}

// out = LeakyReLU([nf | red] @ Wn^T)
__global__ void __launch_bounds__(128) node_apply_wmma(
    const float* __restrict__ nf, const float* __restrict__ red,
    const float* __restrict__ Wn, float* __restrict__ out,
    int N, int numTiles)
{
    __shared__ float sWn[OUT_DIM * KNP];  // ~33 KB
    __shared__ float sA[16 * KNP];        // ~8 KB

    const int tid  = threadIdx.x;
    const int lane = tid & 31;
    const int wave = tid >> 5;

    for (int i = tid; i < OUT_DIM * KN; i += 128) {
        int r = i / KN, c = i - r * KN;
        sWn[r * KNP + c] = Wn[i];
    }

    const int row   = lane & 15;
    const int kb    = (lane >> 4) * 2;
    const int col   = (lane & 15) + wave * 16;
    const int mbase = (lane >> 4) * 8;

    for (int tile = blockIdx.x; tile < numTiles; tile += gridDim.x) {
        const int n0 = tile * 16;
        __syncthreads();
        for (int i = tid; i < 16 * KN; i += 128) {
            int r = i / KN, c = i - r * KN;
            int n = n0 + r;
            float v = 0.0f;
            if (n < N)
                v = (c < IN_DIM) ? nf[n * IN_DIM + c]
                                 : red[n * OUT_DIM + (c - IN_DIM)];
            sA[r * KNP + c] = v;
        }
        __syncthreads();

        v8f acc = {0.f, 0.f, 0.f, 0.f, 0.f, 0.f, 0.f, 0.f};
        #pragma unroll
        for (int k = 0; k < KN; k += 4) {
            v2f a, b;
            a.x = sA[row * KNP + k + kb];
            a.y = sA[row * KNP + k + kb + 1];
            b.x = sWn[col * KNP + k + kb];
            b.y = sWn[col * KNP + k + kb + 1];
            acc = __builtin_amdgcn_wmma_f32_16x16x4_f32(
                false, a, false, b, (short)0, acc, false, false);
        }

        #pragma unroll
        for (int i = 0; i < 8; ++i) {
            int node = n0 + mbase + i;
            if (node < N) out[node * OUT_DIM + col] = leaky(acc[i]);
        }
    }
}

extern "C" void kernel_launch(void* const* d_in, const int* in_sizes, int n_in,
                              void* d_out, int out_size, void* d_ws, size_t ws_size,
                              hipStream_t stream) {
    const float* nf  = (const float*)d_in[0];
    const float* ef  = (const float*)d_in[1];
    const int*   src = (const int*)d_in[2];
    const int*   dst = (const int*)d_in[3];
    const float* We  = (const float*)d_in[4];
    const float* Wn  = (const float*)d_in[5];
    float* out = (float*)d_out;

    const int N = in_sizes[0] / IN_DIM;   // 100000
    const int E = in_sizes[1] / EF_DIM;   // 500000
    float* red = (float*)d_ws;            // N*OUT_DIM floats (25.6 MB)

    zero_f32<<<512, 256, 0, stream>>>(red, N * OUT_DIM);

    const int edgeTiles = (2 * E + 15) / 16;
    const int gridE = edgeTiles < 4096 ? edgeTiles : 4096;
    edge_msg_wmma<<<gridE, 128, 0, stream>>>(nf, ef, src, dst, We, red, E, edgeTiles);

    const int nodeTiles = (N + 15) / 16;
    const int gridN = nodeTiles < 2048 ? nodeTiles : 2048;
    node_apply_wmma<<<gridN, 128, 0, stream>>>(nf, red, Wn, out, N, nodeTiles);
}